// FourierButterworthHFCFilter_9732395893274
// MI455X (gfx1250) — compile-verified
//
#include <hip/hip_runtime.h>
#include <hip/hip_bf16.h>
#include <math.h>

// ---------------------------------------------------------------------------
// FourierButterworthHFCFilter for MI455X (gfx1250, wave32, WMMA)
//
// Pipeline (all on `stream`):
//   1) rank_select (mode raw)  : per-plane lower median of x            -> med[48]
//   2) fft pass 1 (rows, fwd)  : median-fill fused, real->complex       -> A (transposed)
//   3) fft pass 2 (rows, fwd)  : == column FFT; fused Butterworth mul   -> B (natural freq)
//   4) fft pass 3 (rows, inv)                                            -> A
//   5) fft pass 4 (rows, inv)  : fused |.|                               -> hfc (natural)
//   6) rank_select x4 (trunc)  : order stats for 3% / 97% percentiles
//   7) finalize                : (hfc-lo)/(hi-lo)*mask                   -> out
//
// The 512-point FFT is a 16x32 four-step: stage1 = F16(16x16) * X(16x32)
// and stage2 = Y(16x32) * G32(32x32), both mapped onto chained
// V_WMMA_F32_16X16X4_F32 (complex arithmetic = 4 real products).
// ---------------------------------------------------------------------------

typedef float v2f __attribute__((ext_vector_type(2)));
typedef float v8f __attribute__((ext_vector_type(8)));

#define HW      512
#define NPLANE  48            // B*C = 16*3
#define NPIX    262144        // 512*512
#define TWO_PI  6.283185307179586476925f

__device__ __forceinline__ v8f wmma4(v2f a, v2f b, v8f c) {
  // D(16x16,f32) = A(16x4,f32) x B(4x16,f32) + C
  return __builtin_amdgcn_wmma_f32_16x16x4_f32(
      /*neg_a=*/false, a, /*neg_b=*/false, b,
      /*c_mod=*/(short)0, c, /*reuse_a=*/false, /*reuse_b=*/false);
}

// ---------------------------------------------------------------------------
// One wave32 performs a 512-point complex FFT on `row` (512 float2 in LDS),
// in place.  sgn = -1 forward, +1 inverse.  scale applied at the end.
//
// Layouts (per CDNA5 ISA 7.12.2, wave32):
//  A 16x4 f32 : lanes 0-15 M=lane  (v0:K0, v1:K1); lanes16-31 M=lane-16 (K2,K3)
//  B 4x16 f32 : lanes 0-15 N=lane  (v0:K0, v1:K1); lanes16-31 N=lane-16 (K2,K3)
//  C/D 16x16  : vgpr v, lanes 0-15 -> (M=v,   N=lane)
//                        lanes16-31-> (M=v+8, N=lane-16)
// ---------------------------------------------------------------------------
__device__ __forceinline__ void fft512_row(float2* __restrict__ row,
                                           float sgn, float scale, int lane)
{
  const int lh = lane >> 4;   // which half of the wave
  const int ll = lane & 15;

  // Constant A-fragments of the 16-point DFT matrix F16[k1][n1]
  v2f Far[4], Fai[4], FaiN[4];
#pragma unroll
  for (int q = 0; q < 4; ++q) {
#pragma unroll
    for (int c = 0; c < 2; ++c) {
      const int n1 = 4 * q + 2 * lh + c;
      const float ang = sgn * TWO_PI * (float)((ll * n1) & 15) * (1.0f / 16.0f);
      float s_, c_;
      __sincosf(ang, &s_, &c_);
      Far[q][c] = c_; Fai[q][c] = s_; FaiN[q][c] = -s_;
    }
  }

  // -------- Stage 1: Y[k1][n2] = sum_n1 F16[k1][n1] * X[n1][n2]  (x = row[n1*32+n2])
  // then twiddle W512^{k1*n2}; Y stored back to LDS at [k1*32 + n2].
#pragma unroll
  for (int t = 0; t < 2; ++t) {          // n2 tile: [16t, 16t+16)
    v8f yr = {}, yi = {};
    const int n2 = 16 * t + ll;
#pragma unroll
    for (int q = 0; q < 4; ++q) {
      v2f xr, xi;
#pragma unroll
      for (int c = 0; c < 2; ++c) {
        const int n1 = 4 * q + 2 * lh + c;
        const float2 v = row[n1 * 32 + n2];
        xr[c] = v.x; xi[c] = v.y;
      }
      yr = wmma4(Far[q], xr, yr);
      yr = wmma4(FaiN[q], xi, yr);
      yi = wmma4(Far[q], xi, yi);
      yi = wmma4(Fai[q], xr, yi);
    }
#pragma unroll
    for (int v = 0; v < 8; ++v) {
      const int k1 = v + 8 * lh;
      const float ang = sgn * TWO_PI * (float)((k1 * n2) & 511) * (1.0f / 512.0f);
      float s_, c_;
      __sincosf(ang, &s_, &c_);
      const float a = yr[v], b = yi[v];
      row[k1 * 32 + n2] = make_float2(a * c_ - b * s_, a * s_ + b * c_);
    }
  }
  __builtin_amdgcn_wave_barrier();   // keep wave LDS traffic ordered across stages

  // -------- Stage 2: Z[k1][k2] = sum_n2 Y[k1][n2] * G32[n2][k2]
  v8f zr[2], zi[2];
#pragma unroll
  for (int t2 = 0; t2 < 2; ++t2) {       // k2 tile
    v8f accr = {}, acci = {};
    const int k2 = 16 * t2 + ll;
#pragma unroll
    for (int q = 0; q < 8; ++q) {
      v2f ar, ai, gr, gi, giN;
#pragma unroll
      for (int c = 0; c < 2; ++c) {
        const int n2 = 4 * q + 2 * lh + c;
        const float2 v = row[ll * 32 + n2];        // A frag: M=k1=ll, K=n2
        ar[c] = v.x; ai[c] = v.y;
        const float ang = sgn * TWO_PI * (float)((n2 * k2) & 31) * (1.0f / 32.0f);
        float s_, c2_;
        __sincosf(ang, &s_, &c2_);                 // B frag: K=n2, N=k2
        gr[c] = c2_; gi[c] = s_; giN[c] = -s_;
      }
      accr = wmma4(ar, gr, accr);
      accr = wmma4(ai, giN, accr);
      acci = wmma4(ar, gi, acci);
      acci = wmma4(ai, gr, acci);
    }
    zr[t2] = accr; zi[t2] = acci;
  }
  __builtin_amdgcn_wave_barrier();

  // -------- Write Z at natural frequency index k = k1 + 16*k2 (scaled)
#pragma unroll
  for (int t2 = 0; t2 < 2; ++t2) {
    const int k2 = 16 * t2 + ll;
#pragma unroll
    for (int v = 0; v < 8; ++v) {
      const int k1 = v + 8 * lh;
      row[k1 + 16 * k2] = make_float2(zr[t2][v] * scale, zi[t2][v] * scale);
    }
  }
}

// ---------------------------------------------------------------------------
// One FFT pass over "rows" of the current orientation.  16 rows per block
// (8 waves x 2 rows), output written TRANSPOSED with 128B-contiguous stores,
// so chaining 2 passes = 2-D FFT and orientation returns to natural.
//   in_mode : 0 = complex from src2 ; 1 = real from x/mask/med (median fill)
//   out_mode: 0 = complex ; 1 = complex * filter_map ; 2 = |.| -> dst1
// ---------------------------------------------------------------------------
__global__ __launch_bounds__(256) void fft512_pass_kernel(
    const float2* __restrict__ src2,
    const float* __restrict__ xin,
    const float* __restrict__ maskin,
    const float* __restrict__ med,
    const float* __restrict__ fmap,
    float2* __restrict__ dst2,
    float* __restrict__ dst1,
    int in_mode, int out_mode, float sgn, float scale)
{
  __shared__ float2 tile[16][HW];                  // 64 KB
  const int p     = blockIdx.y;
  const int rbase = blockIdx.x * 16;
  const int tid   = threadIdx.x;
  const size_t plane_off = (size_t)p * NPIX;

  // ---- Phase A: coalesced load of 16 rows into LDS
  if (in_mode == 1) {
    const float mv = med[p] + 0.2f;                // median_padding offset
    for (int it = 0; it < 32; ++it) {
      const int idx = it * 256 + tid;
      const int rr = idx >> 9, cc = idx & 511;
      const size_t g = plane_off + (size_t)(rbase + rr) * HW + cc;
      const float xv = xin[g], mk = maskin[g];
      tile[rr][cc] = make_float2(mk * xv + (1.0f - mk) * mv, 0.0f);
    }
  } else {
    for (int it = 0; it < 32; ++it) {
      const int idx = it * 256 + tid;
      const int rr = idx >> 9, cc = idx & 511;
      tile[rr][cc] = src2[plane_off + (size_t)(rbase + rr) * HW + cc];
    }
  }
  __syncthreads();

  // ---- Phase B: each wave runs two 512-point FFTs (EXEC all ones, no divergence)
  const int wave = tid >> 5, lane = tid & 31;
  fft512_row(&tile[wave * 2 + 0][0], sgn, scale, lane);
  fft512_row(&tile[wave * 2 + 1][0], sgn, scale, lane);
  __syncthreads();

  // ---- Phase C: transposed store, 16 consecutive "rows" -> contiguous 128B
  const int rl = tid & 15, kk = tid >> 4;
  for (int it = 0; it < 32; ++it) {
    const int k = it * 16 + kk;
    const float2 v = tile[rl][k];
    const size_t g = plane_off + (size_t)k * HW + (rbase + rl);
    if (out_mode == 0) {
      dst2[g] = v;
    } else if (out_mode == 1) {
      const float f = fmap[(size_t)k * HW + (rbase + rl)];
      dst2[g] = make_float2(v.x * f, v.y * f);
    } else {
      dst1[g] = sqrtf(v.x * v.x + v.y * v.y);
    }
  }
}

// ---------------------------------------------------------------------------
// Exact order statistic per plane via 4-pass 256-bin radix select on
// order-preserving float->uint keys.  mode 1 applies trunc(f*256)/256 first.
// ---------------------------------------------------------------------------
__device__ __forceinline__ unsigned f2key(float f) {
  const unsigned u = __float_as_uint(f);
  return (u & 0x80000000u) ? ~u : (u | 0x80000000u);
}
__device__ __forceinline__ float key2f(unsigned k) {
  const unsigned u = (k & 0x80000000u) ? (k & 0x7FFFFFFFu) : ~k;
  return __uint_as_float(u);
}

__global__ __launch_bounds__(256) void rank_select_kernel(
    const float* __restrict__ src, float* __restrict__ dst, int rank, int mode)
{
  __shared__ unsigned cnt[256];
  __shared__ unsigned s_prefix;
  __shared__ int s_rank;
  const int p = blockIdx.x;
  const int tid = threadIdx.x;
  const size_t base = (size_t)p * NPIX;
  if (tid == 0) { s_prefix = 0u; s_rank = rank; }

  for (int pass = 0; pass < 4; ++pass) {
    cnt[tid] = 0u;
    __syncthreads();
    const int shift = 24 - 8 * pass;
    const unsigned kmask = (pass == 0) ? 0u : (0xFFFFFFFFu << (shift + 8));
    const unsigned prefix = s_prefix;
    for (int i = tid; i < NPIX; i += 256) {
      float f = src[base + i];
      if (mode) f = truncf(f * 256.0f) * (1.0f / 256.0f);
      const unsigned k = f2key(f);
      if ((k & kmask) == prefix)
        atomicAdd(&cnt[(k >> shift) & 255u], 1u);
    }
    __syncthreads();
    if (tid == 0) {
      int r = s_rank;
      unsigned cum = 0;
      int b = 255;
      for (int i2 = 0; i2 < 256; ++i2) {
        const unsigned c = cnt[i2];
        if (cum + c > (unsigned)r) { b = i2; break; }
        cum += c;
      }
      s_rank = r - (int)cum;
      s_prefix = prefix | ((unsigned)b << shift);
    }
    __syncthreads();
  }
  if (tid == 0) dst[p] = key2f(s_prefix);
}

// ---------------------------------------------------------------------------
// out = (hfc - lo) / (hi - lo) * mask, lo/hi interpolated from order stats
// ---------------------------------------------------------------------------
__global__ __launch_bounds__(256) void finalize_kernel(
    const float* __restrict__ hfc, const float* __restrict__ mask,
    const float* __restrict__ q,   float* __restrict__ out,
    float frac_lo, float frac_hi)
{
  const size_t i = (size_t)blockIdx.x * 256 + threadIdx.x;
  const int p = (int)(i >> 18);                 // / NPIX
  const float lo = q[p]       + frac_lo * (q[NPLANE + p]     - q[p]);
  const float hi = q[2*NPLANE + p] + frac_hi * (q[3*NPLANE + p] - q[2*NPLANE + p]);
  out[i] = (hfc[i] - lo) / (hi - lo) * mask[i];
}

// ---------------------------------------------------------------------------
extern "C" void kernel_launch(void* const* d_in, const int* in_sizes, int n_in,
                              void* d_out, int out_size, void* d_ws, size_t ws_size,
                              hipStream_t stream) {
  const float* x    = (const float*)d_in[0];     // (16,3,512,512)
  const float* mask = (const float*)d_in[1];     // (16,3,512,512)
  const float* fm   = (const float*)d_in[2];     // (512,512)
  float* out = (float*)d_out;

  char* ws = (char*)d_ws;
  const size_t BUF = (size_t)NPLANE * NPIX * sizeof(float2);   // 100,663,296 B
  float2* A   = (float2*)ws;               // ping
  float2* Bb  = (float2*)(ws + BUF);       // pong
  float*  hfc = (float*)Bb;                // |ifft2| reuses pong (dead by then)
  float*  st  = (float*)(ws + 2 * BUF);    // med[48], q0..q3[4*48]
  float*  med = st;
  float*  q   = st + NPLANE;

  // percentile positions: idx = pct/100 * (NPIX-1)
  const double flo = 0.03 * (double)(NPIX - 1);   // 7864.29
  const double fhi = 0.97 * (double)(NPIX - 1);   // 254278.71
  const int lo0 = (int)flo, hi0 = (int)fhi;
  const float frac_lo = (float)(flo - (double)lo0);
  const float frac_hi = (float)(fhi - (double)hi0);
  const int med_rank = (NPIX - 1) / 2;            // torch lower median

  const dim3 fgrid(HW / 16, NPLANE);
  const float inv512 = 1.0f / 512.0f;

  // 1) per-plane median of x
  rank_select_kernel<<<NPLANE, 256, 0, stream>>>(x, med, med_rank, 0);

  // 2) rows forward (fused median fill), write transposed -> A
  fft512_pass_kernel<<<fgrid, 256, 0, stream>>>(nullptr, x, mask, med, nullptr,
                                                A, nullptr, 1, 0, -1.0f, 1.0f);
  // 3) cols forward + Butterworth filter, write transposed -> B (natural freq)
  fft512_pass_kernel<<<fgrid, 256, 0, stream>>>(A, nullptr, nullptr, nullptr, fm,
                                                Bb, nullptr, 0, 1, -1.0f, 1.0f);
  // 4) rows inverse -> A
  fft512_pass_kernel<<<fgrid, 256, 0, stream>>>(Bb, nullptr, nullptr, nullptr, nullptr,
                                                A, nullptr, 0, 0, 1.0f, inv512);
  // 5) cols inverse + |.| -> hfc (natural orientation)
  fft512_pass_kernel<<<fgrid, 256, 0, stream>>>(A, nullptr, nullptr, nullptr, nullptr,
                                                nullptr, hfc, 0, 2, 1.0f, inv512);

  // 6) percentile order statistics on trunc(hfc*256)/256
  rank_select_kernel<<<NPLANE, 256, 0, stream>>>(hfc, q + 0 * NPLANE, lo0,     1);
  rank_select_kernel<<<NPLANE, 256, 0, stream>>>(hfc, q + 1 * NPLANE, lo0 + 1, 1);
  rank_select_kernel<<<NPLANE, 256, 0, stream>>>(hfc, q + 2 * NPLANE, hi0,     1);
  rank_select_kernel<<<NPLANE, 256, 0, stream>>>(hfc, q + 3 * NPLANE, hi0 + 1, 1);

  // 7) normalize + re-mask
  const int total_blocks = (NPLANE * NPIX) / 256;
  finalize_kernel<<<total_blocks, 256, 0, stream>>>(hfc, mask, q, out,
                                                    frac_lo, frac_hi);
}